// GAT_47940424958478
// MI455X (gfx1250) — compile-verified
//
#include <hip/hip_runtime.h>
#include <hip/hip_bf16.h>

typedef __attribute__((ext_vector_type(2))) float v2f;
typedef __attribute__((ext_vector_type(8))) float v8f;

#define NN 50000
#define EE 800000
#define FF 128
#define CCH 64
#define GG 256
#define ETOT (EE + NN)
#define NEG_SLOPE 0.2f

// ---------------------------------------------------------------------------
// GEMM: Y[m, n] = sum_k A[m, k] * W[n, k] + bias[n]   (W row-major [64, K])
// One 16-row slab per block; 4 waves -> 4 16x16 column tiles (64 cols).
// M (rows) must be a multiple of 16 (50000 and 256 both are).
// Uses V_WMMA_F32_16X16X4_F32; fragment layouts per CDNA5 ISA 7.12.2:
//   A (16x4 f32):  VGPR0 = K=0 | K=2 (lane halves), VGPR1 = K=1 | K=3
//   B (4x16 f32):  same K split, N striped over lanes 0-15
//   D (16x16 f32): VGPR r -> M=r (lanes 0-15), M=r+8 (lanes 16-31)
// ---------------------------------------------------------------------------
__global__ __launch_bounds__(128)
void gemm_wmma_f32(const float* __restrict__ A, int lda,
                   const float* __restrict__ W,
                   const float* __restrict__ bias,
                   float* __restrict__ Y, int ldy,
                   int K, int do_relu)
{
  const int wave = threadIdx.x >> 5;
  const int lane = threadIdx.x & 31;
  const int half = lane >> 4;      // lane group 0-15 vs 16-31
  const int l16  = lane & 15;
  const int rowBase = blockIdx.x << 4;
  const int colBase = wave << 4;

  const float* arow = A + (size_t)(rowBase + l16) * lda;
  const float* wrow = W + (size_t)(colBase + l16) * K;

  v8f d = {};
  for (int k = 0; k < K; k += 4) {
    const int ka = k + 2 * half;
    v2f a, b;
    a.x = arow[ka];
    a.y = arow[ka + 1];
    b.x = wrow[ka];
    b.y = wrow[ka + 1];
#if __has_builtin(__builtin_amdgcn_wmma_f32_16x16x4_f32)
    d = __builtin_amdgcn_wmma_f32_16x16x4_f32(false, a, false, b,
                                              (short)0, d, false, false);
#else
    // scalar fallback (should not be taken on gfx1250; histogram will show wmma=0)
    #pragma unroll
    for (int r = 0; r < 8; ++r) {
      d[r] += a.x * b.x + a.y * b.y;  // placeholder so code still compiles
    }
#endif
  }

  const int col = colBase + l16;
  const float bv = bias[col];
  #pragma unroll
  for (int r = 0; r < 8; ++r) {
    const int row = rowBase + r + 8 * half;
    float v = d[r] + bv;
    if (do_relu) v = fmaxf(v, 0.0f);
    Y[(size_t)row * ldy + col] = v;
  }
}

// ---------------------------------------------------------------------------
// Edge kernels: warp-per-edge, lane = channel (64 ch / wave32 = 2 per lane).
// xl/xr are dense N x 64 fp32 buffers (12.8 MB each -> resident in 192MB L2),
// so the random src/dst gathers are L2 hits; lanes read consecutive channels
// -> coalesced 256B row reads.
// ---------------------------------------------------------------------------
__device__ __forceinline__ void edge_endpoints(const int* __restrict__ ei,
                                               int e, int& src, int& dst)
{
  if (e < EE) { src = ei[e]; dst = ei[EE + e]; }
  else        { src = e - EE; dst = e - EE; }   // appended self-loops
}

__device__ __forceinline__ unsigned enc_f32(float f)
{
  unsigned u = __float_as_uint(f);
  return (u & 0x80000000u) ? ~u : (u | 0x80000000u);
}
__device__ __forceinline__ float dec_f32(unsigned e)
{
  return __uint_as_float((e & 0x80000000u) ? (e ^ 0x80000000u) : ~e);
}

__global__ __launch_bounds__(256)
void edge_score_max(const int* __restrict__ ei,
                    const float* __restrict__ xl,
                    const float* __restrict__ xr,
                    const float* __restrict__ att,
                    float* __restrict__ escore,
                    unsigned* __restrict__ menc)
{
  const int e = blockIdx.x * 8 + (threadIdx.x >> 5);
  if (e >= ETOT) return;
  const int lane = threadIdx.x & 31;
  int src, dst;
  edge_endpoints(ei, e, src, dst);
  const float* pl = xl + (size_t)src * CCH;
  const float* pr = xr + (size_t)dst * CCH;
  float acc = 0.0f;
  #pragma unroll
  for (int h = 0; h < 2; ++h) {
    const int ch = lane + 32 * h;
    float s = pl[ch] + pr[ch];
    s = (s > 0.0f) ? s : NEG_SLOPE * s;     // leaky_relu
    acc += s * att[ch];
  }
  #pragma unroll
  for (int off = 16; off > 0; off >>= 1)
    acc += __shfl_down(acc, off, 32);
  if (lane == 0) {
    escore[e] = acc;
    atomicMax(&menc[dst], enc_f32(acc));    // order-preserving encoding
  }
}

__global__ __launch_bounds__(256)
void edge_expsum(const int* __restrict__ ei,
                 float* __restrict__ escore,
                 const unsigned* __restrict__ menc,
                 float* __restrict__ ssum)
{
  const int e = blockIdx.x * 256 + threadIdx.x;
  if (e >= ETOT) return;
  const int dst = (e < EE) ? ei[EE + e] : (e - EE);
  const float m  = dec_f32(menc[dst]);
  const float ex = __expf(escore[e] - m);
  escore[e] = ex;
  atomicAdd(&ssum[dst], ex);
}

__global__ __launch_bounds__(256)
void edge_scatter(const int* __restrict__ ei,
                  const float* __restrict__ escore,
                  const float* __restrict__ ssum,
                  const float* __restrict__ xl,
                  float* __restrict__ acc)
{
  const int e = blockIdx.x * 8 + (threadIdx.x >> 5);
  if (e >= ETOT) return;
  const int lane = threadIdx.x & 31;
  int src, dst;
  edge_endpoints(ei, e, src, dst);
  const float alpha = escore[e] / (ssum[dst] + 1e-16f);
  const float* pl = xl + (size_t)src * CCH;
  float*       pd = acc + (size_t)dst * CCH;
  #pragma unroll
  for (int h = 0; h < 2; ++h) {
    const int ch = lane + 32 * h;
    atomicAdd(&pd[ch], alpha * pl[ch]);     // global_atomic_add_f32
  }
}

// out[n, c] = relu(acc[n, c] + bo[c]) written into xcat column block (ld=192)
__global__ __launch_bounds__(256)
void bias_relu_store(const float* __restrict__ acc,
                     const float* __restrict__ bo,
                     float* __restrict__ dstcol, int ldd)
{
  const int i = blockIdx.x * 256 + threadIdx.x;
  if (i >= NN * CCH) return;
  const int n = i >> 6, c = i & 63;
  const float v = acc[i] + bo[c];
  dstcol[(size_t)n * ldd + c] = fmaxf(v, 0.0f);
}

__global__ __launch_bounds__(256)
void pool_add(const float* __restrict__ h,
              const int* __restrict__ batch,
              float* __restrict__ g)
{
  const int i = blockIdx.x * 256 + threadIdx.x;
  if (i >= NN * CCH) return;
  const int n = i >> 6, c = i & 63;
  atomicAdd(&g[(size_t)batch[n] * CCH + c], h[i]);
}

__global__ __launch_bounds__(256)
void cls_head(const float* __restrict__ emb,
              const float* __restrict__ w,
              const float* __restrict__ b,
              float* __restrict__ out)
{
  const int i = blockIdx.x * 256 + threadIdx.x;   // i = g*2 + j
  if (i >= GG * 2) return;
  const int gi = i >> 1, j = i & 1;
  float acc = b[j];
  #pragma unroll 8
  for (int c = 0; c < CCH; ++c)
    acc += emb[gi * CCH + c] * w[j * CCH + c];
  out[i] = acc;
}

// ---------------------------------------------------------------------------
extern "C" void kernel_launch(void* const* d_in, const int* in_sizes, int n_in,
                              void* d_out, int out_size, void* d_ws, size_t ws_size,
                              hipStream_t stream)
{
  (void)in_sizes; (void)n_in; (void)out_size; (void)ws_size;

  const float* x     = (const float*)d_in[0];
  const int*   ei    = (const int*)  d_in[1];
  const int*   batch = (const int*)  d_in[2];
  const float* wl[3]  = {(const float*)d_in[3],  (const float*)d_in[9],  (const float*)d_in[15]};
  const float* bl[3]  = {(const float*)d_in[4],  (const float*)d_in[10], (const float*)d_in[16]};
  const float* wr[3]  = {(const float*)d_in[5],  (const float*)d_in[11], (const float*)d_in[17]};
  const float* br[3]  = {(const float*)d_in[6],  (const float*)d_in[12], (const float*)d_in[18]};
  const float* att[3] = {(const float*)d_in[7],  (const float*)d_in[13], (const float*)d_in[19]};
  const float* bo[3]  = {(const float*)d_in[8],  (const float*)d_in[14], (const float*)d_in[20]};
  const float* lin1_w = (const float*)d_in[21];
  const float* lin1_b = (const float*)d_in[22];
  const float* pred_w = (const float*)d_in[23];
  const float* pred_b = (const float*)d_in[24];
  const float* cls_w  = (const float*)d_in[25];
  const float* cls_b  = (const float*)d_in[26];

  // Workspace carve-up (all fp32; ~94 MB total)
  char* wsb = (char*)d_ws;
  auto alloc = [&](size_t nfloats) {
    float* p = (float*)wsb;
    wsb += nfloats * sizeof(float);
    return p;
  };
  float*    xcat   = alloc((size_t)NN * 3 * CCH);  // [x1 | x2 | x3], ld = 192
  float*    xl     = alloc((size_t)NN * CCH);
  float*    xr     = alloc((size_t)NN * CCH);
  float*    accb   = alloc((size_t)NN * CCH);
  float*    escore = alloc((size_t)ETOT);
  unsigned* menc   = (unsigned*)alloc((size_t)NN);
  float*    ssum   = alloc((size_t)NN);
  float*    hbuf   = alloc((size_t)NN * CCH);
  float*    gbuf   = alloc((size_t)GG * CCH);
  float*    embb   = alloc((size_t)GG * CCH);

  const int gemmRowBlocks = NN / 16;                 // 3125
  const int edgeWarpGrid  = (ETOT + 7) / 8;          // warp-per-edge, 8 warps/block
  const int edgeThrGrid   = (ETOT + 255) / 256;
  const int nodeChGrid    = (NN * CCH + 255) / 256;

  for (int L = 0; L < 3; ++L) {
    const float* in  = (L == 0) ? x : (xcat + (size_t)CCH * (L - 1));
    const int    lda = (L == 0) ? FF : 3 * CCH;
    const int    K   = (L == 0) ? FF : CCH;

    gemm_wmma_f32<<<gemmRowBlocks, 128, 0, stream>>>(in, lda, wl[L], bl[L], xl, CCH, K, 0);
    gemm_wmma_f32<<<gemmRowBlocks, 128, 0, stream>>>(in, lda, wr[L], br[L], xr, CCH, K, 0);

    hipMemsetAsync(accb, 0, (size_t)NN * CCH * sizeof(float), stream);
    hipMemsetAsync(menc, 0, (size_t)NN * sizeof(unsigned), stream);  // below all encodings
    hipMemsetAsync(ssum, 0, (size_t)NN * sizeof(float), stream);

    edge_score_max<<<edgeWarpGrid, 256, 0, stream>>>(ei, xl, xr, att[L], escore, menc);
    edge_expsum  <<<edgeThrGrid,  256, 0, stream>>>(ei, escore, menc, ssum);
    edge_scatter <<<edgeWarpGrid, 256, 0, stream>>>(ei, escore, ssum, xl, accb);

    bias_relu_store<<<nodeChGrid, 256, 0, stream>>>(accb, bo[L],
                                                    xcat + (size_t)CCH * L, 3 * CCH);
  }

  // h = relu(xcat @ lin1_w.T + lin1_b)
  gemm_wmma_f32<<<gemmRowBlocks, 128, 0, stream>>>(xcat, 3 * CCH, lin1_w, lin1_b,
                                                   hbuf, CCH, 3 * CCH, 1);
  // g = segment_sum(h, batch)
  hipMemsetAsync(gbuf, 0, (size_t)GG * CCH * sizeof(float), stream);
  pool_add<<<nodeChGrid, 256, 0, stream>>>(hbuf, batch, gbuf);

  // emb = relu(g @ pred_w.T + pred_b)
  gemm_wmma_f32<<<GG / 16, 128, 0, stream>>>(gbuf, CCH, pred_w, pred_b, embb, CCH, CCH, 1);

  // out = emb @ cls_w.T + cls_b
  cls_head<<<2, 256, 0, stream>>>(embb, cls_w, cls_b, (float*)d_out);
}